// SimpleFractalModel6_4303557230622
// MI455X (gfx1250) — compile-verified
//
#include <hip/hip_runtime.h>

// MI455X / gfx1250, wave32. FP32 WMMA: D(16x16) = A(16x4) x B(4x16) + C.
typedef __attribute__((ext_vector_type(2))) float v2f;
typedef __attribute__((ext_vector_type(8))) float v8f;

#define HID      16
#define NLAYERS  32
#define INDIM    124   // 4 + 4*2*15
#define NCHUNK1  31    // 124 / 4

__device__ __forceinline__ v8f wmma4(v2f a, v2f b, v8f c) {
    // 8 args: (neg_a, A, neg_b, B, c_mod, C, reuse_a, reuse_b)
    return __builtin_amdgcn_wmma_f32_16x16x4_f32(false, a, false, b, (short)0, c,
                                                 false, false);
}

__global__ __launch_bounds__(256)
void fractal_mlp_wmma(const float* __restrict__ z,  const float* __restrict__ c,
                      const float* __restrict__ W1, const float* __restrict__ b1,
                      const float* __restrict__ Ws, const float* __restrict__ bs,
                      const float* __restrict__ Wf, const float* __restrict__ bf,
                      float* __restrict__ out, int B)
{
    // ---- weights staged (transposed) in LDS: ~42.3 KB of 320 KB/WGP ----
    __shared__ float sW1t[16][128];              // sW1t[out][in]  (in padded 124->128)
    __shared__ float sWst[NLAYERS][16][16];      // sWst[l][out][in]
    __shared__ float sB1[16];
    __shared__ float sBs[NLAYERS * 16];
    __shared__ float sWf[16];
    __shared__ float sBf;

    const int tid = threadIdx.x;
    for (int idx = tid; idx < 16 * 128; idx += 256) {
        int mo = idx >> 7, k = idx & 127;
        sW1t[mo][k] = (k < INDIM) ? W1[k * HID + mo] : 0.0f;
    }
    for (int idx = tid; idx < NLAYERS * 256; idx += 256) {
        int l = idx >> 8, r = idx & 255, mo = r >> 4, k = r & 15;
        sWst[l][mo][k] = Ws[l * 256 + k * HID + mo];
    }
    for (int idx = tid; idx < NLAYERS * 16; idx += 256) sBs[idx] = bs[idx];
    if (tid < 16) { sB1[tid] = b1[tid]; sWf[tid] = Wf[tid]; }
    if (tid == 0) sBf = bf[0];
    __syncthreads();

    const int lane  = tid & 31;
    const int wave  = tid >> 5;
    const int hi    = lane >> 4;    // half-wave: selects K sub-pair / hidden M+8
    const int m     = lane & 15;    // row-within-tile / sample-within-tile / N index
    const int sbase = (blockIdx.x * 8 + wave) * 32;   // 32 samples per wave (2 tiles)
    const int s     = sbase + lane;                   // this lane's sample for loads

    float xz0 = 0.f, xz1 = 0.f, xc0 = 0.f, xc1 = 0.f;
    if (s < B) {
        xz0 = z[2 * s]; xz1 = z[2 * s + 1];
        xc0 = c[2 * s]; xc1 = c[2 * s + 1];
    }

    // Per-tile broadcast of the owning lane's x; fold the lane-half component
    // select (d = 2*hi + v) immediately.
    float xa[2], xb[2];
#pragma unroll
    for (int t = 0; t < 2; ++t) {
        const int src = t * 16 + m;
        float x0 = __shfl(xz0, src, 32);
        float x1 = __shfl(xz1, src, 32);
        float x2 = __shfl(xc0, src, 32);
        float x3 = __shfl(xc1, src, 32);
        xa[t] = hi ? x2 : x0;   // x[d] for d = 2*hi + 0
        xb[t] = hi ? x3 : x1;   // x[d] for d = 2*hi + 1
    }

    // hT tiles in WMMA C/D layout: hc[t][r] = h[hidden = r + 8*hi][sample = m]
    float hc[2][8];

    // -------- Layer 1: hT = leaky(W1^T @ featT + b1), K = 124 (two acc chains) -------
    {
        v8f acc0, acc1;
#pragma unroll
        for (int r = 0; r < 8; ++r) { acc0[r] = sB1[r + 8 * hi]; acc1[r] = acc0[r]; }

#pragma unroll
        for (int j = 0; j < NCHUNK1; ++j) {
            // A = W1^T chunk, loaded ONCE and shared by both tiles (b64 LDS read):
            // lane holds A[m][2*hi + v] = W1[4j + 2*hi + v][m]
            v2f a = *(const v2f*)&sW1t[m][4 * j + 2 * hi];
            // B = positional-encoding features, generated in-register, once each.
            v2f b0, b1v;
            if (j == 0) {                            // raw x features (k < 4)
                b0[0] = xa[0]; b0[1] = xb[0];
                b1v[0] = xa[1]; b1v[1] = xb[1];
            } else {
                const float scale = (float)(1u << ((j - 1) >> 1));  // 2^i, compile-time
                if ((j - 1) & 1) {
                    b0[0]  = __cosf(scale * xa[0]); b0[1]  = __cosf(scale * xb[0]);
                    b1v[0] = __cosf(scale * xa[1]); b1v[1] = __cosf(scale * xb[1]);
                } else {
                    b0[0]  = __sinf(scale * xa[0]); b0[1]  = __sinf(scale * xb[0]);
                    b1v[0] = __sinf(scale * xa[1]); b1v[1] = __sinf(scale * xb[1]);
                }
            }
            acc0 = wmma4(a, b0, acc0);               // two independent WMMA chains
            acc1 = wmma4(a, b1v, acc1);
        }
#pragma unroll
        for (int r = 0; r < 8; ++r) {
            float d0 = acc0[r]; hc[0][r] = fmaxf(d0, 0.2f * d0);
            float d1 = acc1[r]; hc[1][r] = fmaxf(d1, 0.2f * d1);
        }
    }

    // ---------------- 32 residual layers: hT += leaky(Wl^T @ hT + bl) ----------------
    for (int l = 0; l < NLAYERS; ++l) {
        v2f aq[4];
#pragma unroll
        for (int q = 0; q < 4; ++q)
            aq[q] = *(const v2f*)&sWst[l][m][4 * q + 2 * hi];   // shared by both tiles
        float bl[8];
#pragma unroll
        for (int r = 0; r < 8; ++r) bl[r] = sBs[l * 16 + r + 8 * hi];

#pragma unroll
        for (int t = 0; t < 2; ++t) {
            v8f acc;
#pragma unroll
            for (int r = 0; r < 8; ++r) acc[r] = bl[r];
#pragma unroll
            for (int q = 0; q < 4; ++q) {
                // B chunk from hT (C-layout). Lanes 0-15 need k_lo = 4q+v, lanes
                // 16-31 need k_hi = 4q+v+2; for q<2 both live in lower-half lanes
                // (src = m), for q>=2 both live in upper-half lanes (src = m+16).
                // So one half of the wave reads its OWN register -> 1 shuffle +
                // 1 select per B register (instead of 2 shuffles).
                v2f b;
#pragma unroll
                for (int v = 0; v < 2; ++v) {
                    const int k_lo = 4 * q + v;
                    const int k_hi = k_lo + 2;
                    if (q < 2) {
                        float o = __shfl(hc[t][k_hi & 7], m, 32);        // for lanes 16-31
                        b[v] = hi ? o : hc[t][k_lo & 7];                 // lanes 0-15: own reg
                    } else {
                        float o = __shfl(hc[t][k_lo & 7], m + 16, 32);   // for lanes 0-15
                        b[v] = hi ? hc[t][k_hi & 7] : o;                 // lanes 16-31: own reg
                    }
                }
                acc = wmma4(aq[q], b, acc);
            }
#pragma unroll
            for (int r = 0; r < 8; ++r) {
                float d = acc[r];
                hc[t][r] += fmaxf(d, 0.2f * d);      // residual + LeakyReLU(0.2)
            }
        }
    }

    // ---------------- Head: out[n] = sum_m hT[m][n] * Wf[m] + bf ----------------
    float wfr[8];
#pragma unroll
    for (int r = 0; r < 8; ++r) wfr[r] = sWf[r + 8 * hi];
    const float bfv = sBf;
#pragma unroll
    for (int t = 0; t < 2; ++t) {
        float p = 0.f;
#pragma unroll
        for (int r = 0; r < 8; ++r) p += hc[t][r] * wfr[r];
        p += __shfl_xor(p, 16, 32);                  // combine the two hidden halves
        const int so = sbase + t * 16 + m;
        if (hi == 0 && so < B) out[so] = p + bfv;
    }
}

extern "C" void kernel_launch(void* const* d_in, const int* in_sizes, int n_in,
                              void* d_out, int out_size, void* d_ws, size_t ws_size,
                              hipStream_t stream) {
    const float* z  = (const float*)d_in[0];
    const float* c  = (const float*)d_in[1];
    const float* W1 = (const float*)d_in[2];
    const float* b1 = (const float*)d_in[3];
    const float* Ws = (const float*)d_in[4];
    const float* bs = (const float*)d_in[5];
    const float* Wf = (const float*)d_in[6];
    const float* bf = (const float*)d_in[7];
    float* out = (float*)d_out;

    const int B = in_sizes[0] / 2;           // z is (B, 2)
    const int blocks = (B + 255) / 256;      // 8 waves/block, 32 samples/wave
    fractal_mlp_wmma<<<blocks, 256, 0, stream>>>(z, c, W1, b1, Ws, bs, Wf, bf, out, B);
}